// MiniBatchDiscrimination_60653528154297
// MI455X (gfx1250) — compile-verified
//
#include <hip/hip_runtime.h>

// Problem constants (from the reference)
#define BATCH    512
#define IN_FEAT  512
#define OUT_FEAT 64
#define KDIM     32
#define NCOL     (OUT_FEAT * KDIM)     // 2048 : columns of M
#define OUTROW   (IN_FEAT + OUT_FEAT)  // 576  : output row stride

typedef __attribute__((ext_vector_type(2))) float v2f;
typedef __attribute__((ext_vector_type(8))) float v8f;

// ---------------------------------------------------------------------------
// Kernel 1: M = x @ W  (W = T viewed as (IN_FEAT, NCOL) row-major)
// f32 WMMA 16x16x4.  One wave per 16x16 output tile, 4 waves per block.
// Grid: (NCOL/64, BATCH/16), block 128 threads.
// ---------------------------------------------------------------------------
__global__ __launch_bounds__(128)
void mbd_gemm_wmma_f32(const float* __restrict__ X,
                       const float* __restrict__ W,
                       float* __restrict__ M)
{
    const int lane = threadIdx.x & 31;
    const int wave = threadIdx.x >> 5;        // 0..3
    const int half = lane >> 4;               // 0 or 1
    const int l16  = lane & 15;

    const int m0 = blockIdx.y << 4;                    // row tile base
    const int n0 = (blockIdx.x << 6) + (wave << 4);    // col tile base

    // A 16x4 f32 layout: lanes 0-15 -> K={0,1}, lanes 16-31 -> K={2,3};
    //   per-lane VGPR0/1 hold consecutive K.  Row M = lane&15.
    // B 4x16 f32 layout (mirror): N = lane&15, per-lane K = {2*half, 2*half+1}.
    const float* xrow = X + (m0 + l16) * IN_FEAT;  // A source row
    const float* wcol = W + n0 + l16;              // B source column

    v8f acc = {};

    #pragma unroll 4
    for (int k = 0; k < IN_FEAT; k += 4) {
        v2f a, b;
        const float2 av = *(const float2*)(xrow + k + 2 * half);
        a.x = av.x;
        a.y = av.y;
        b.x = wcol[(k + 2 * half)     * NCOL];
        b.y = wcol[(k + 2 * half + 1) * NCOL];
        // D = A*B + C   (8 args: neg_a, A, neg_b, B, c_mod, C, reuse_a, reuse_b)
        acc = __builtin_amdgcn_wmma_f32_16x16x4_f32(
                  false, a, false, b, (short)0, acc, false, false);
    }

    // C/D layout: VGPR r -> M = r + 8*half, N = lane&15
    float* mout = M + n0 + l16;
    #pragma unroll
    for (int r = 0; r < 8; ++r)
        mout[(m0 + r + 8 * half) * NCOL] = acc[r];
}

// ---------------------------------------------------------------------------
// Kernel 2: mbd[b,o] = sum_{b'} exp(-sum_k |M[b,o,k]-M[b',o,k]|) - 1
// Block = 256 threads = 4 groups x 64; group g owns row b_j = 4*blk + g,
// thread (g,o) keeps M[b_j,o,0..31] in registers.  b_i rows streamed in
// tiles of 4 through LDS (padded stride to dodge bank conflicts).
// Also copies x into out[:, 0:512].
// ---------------------------------------------------------------------------
#define BJ   4
#define TI   4
#define OPAD 36   // padded floats per o-slot (36%4==0 -> float4-aligned stores)

__global__ __launch_bounds__(256)
void mbd_pairwise_kernel(const float* __restrict__ X,
                         const float* __restrict__ M,
                         float* __restrict__ out)
{
    __shared__ float tile[TI][OUT_FEAT * OPAD];   // 4*64*36*4B = 36 KB

    const int tid   = threadIdx.x;     // 0..255
    const int g     = tid >> 6;        // 0..3
    const int o     = tid & 63;        // 0..63
    const int jbase = blockIdx.x * BJ;

    // Own row fragment in registers: M[jbase+g][o][0..31]
    float mj[KDIM];
    {
        const float* mrow = M + (jbase + g) * NCOL + o * KDIM;
        #pragma unroll
        for (int t = 0; t < KDIM / 4; ++t) {
            float4 v = ((const float4*)mrow)[t];
            mj[4 * t + 0] = v.x; mj[4 * t + 1] = v.y;
            mj[4 * t + 2] = v.z; mj[4 * t + 3] = v.w;
        }
    }

    float acc = 0.0f;

    for (int i0 = 0; i0 < BATCH; i0 += TI) {
        __syncthreads();   // protect LDS from previous iteration's readers
        // Stage TI rows: each thread moves 8 contiguous floats per row,
        // landing inside a single o-slot (k0 = (tid&3)*8).
        #pragma unroll
        for (int ti = 0; ti < TI; ++ti) {
            const float* src = M + (i0 + ti) * NCOL + tid * 8;
            float4 v0 = ((const float4*)src)[0];
            float4 v1 = ((const float4*)src)[1];
            float* dst = &tile[ti][(tid >> 2) * OPAD + (tid & 3) * 8];
            ((float4*)dst)[0] = v0;
            ((float4*)dst)[1] = v1;
        }
        __syncthreads();

        #pragma unroll
        for (int ti = 0; ti < TI; ++ti) {
            const float* mi = &tile[ti][o * OPAD];
            float s = 0.0f;
            #pragma unroll
            for (int kk = 0; kk < KDIM; ++kk)
                s += fabsf(mj[kk] - mi[kk]);   // v_add_f32 with |src| modifier
            acc += __expf(-s);                 // v_exp_f32 path
        }
    }

    // Self pair (b_i == b_j) contributed exp(0)=1; reference subtracts it.
    out[(jbase + g) * OUTROW + IN_FEAT + o] = acc - 1.0f;

    // Epilogue: copy x rows into out[:, 0:IN_FEAT]
    for (int idx = tid; idx < BJ * IN_FEAT; idx += 256) {
        int r = idx >> 9;          // /512
        int c = idx & (IN_FEAT - 1);
        out[(jbase + r) * OUTROW + c] = X[(jbase + r) * IN_FEAT + c];
    }
}

// ---------------------------------------------------------------------------
extern "C" void kernel_launch(void* const* d_in, const int* in_sizes, int n_in,
                              void* d_out, int out_size, void* d_ws, size_t ws_size,
                              hipStream_t stream)
{
    (void)in_sizes; (void)n_in; (void)out_size; (void)ws_size;

    const float* x = (const float*)d_in[0];   // (512, 512) f32
    const float* T = (const float*)d_in[1];   // (512, 64, 32) f32 == (512, 2048)
    float* out = (float*)d_out;               // (512, 576) f32
    float* M   = (float*)d_ws;                // scratch: 512*2048*4 = 4 MB

    dim3 gemm_grid(NCOL / 64, BATCH / 16);    // (32, 32)
    mbd_gemm_wmma_f32<<<gemm_grid, 128, 0, stream>>>(x, T, M);

    mbd_pairwise_kernel<<<BATCH / BJ, 256, 0, stream>>>(x, M, out);
}